// AudioLSTM_83562883711342
// MI455X (gfx1250) — compile-verified
//
#include <hip/hip_runtime.h>
#include <hip/hip_bf16.h>

// ---------------------------------------------------------------------------
// 2-layer bidirectional LSTM (B=64, T=1500, D_IN=80, H=128) for gfx1250.
//
//   Phase 1 (per layer): pre[d, m, g] = in[m,:] . Wih[d,g,:] + bih + bhh
//       -> big parallel GEMM, M = B*T = 96000, N = 4H = 512, K = 80 / 256
//       -> v_wmma_f32_16x16x4_f32, thousands of workgroups.
//   Phase 2 (per layer): serial scan over T. Batch is independent, so
//       grid = (4 batch-tiles of 16) x (2 directions) persistent workgroups,
//       1024 threads = 32 waves, wave j owns gate columns [16j, 16j+16).
//       Whh^T fragments live in registers for all 1500 steps; h/c/g in LDS.
//       pre tile is double-buffered in registers + prefetched into L2;
//       activations use the CDNA5 V_TANH_F32 transcendental.
//
// Workspace layout (requires ~470 MB):
//   pre : 2 * 96000 * 512 f32   (reused by both layers)
//   out0: 96000 * 256 f32       (layer-0 bidirectional output)
// ---------------------------------------------------------------------------

typedef __attribute__((ext_vector_type(2))) float v2f;
typedef __attribute__((ext_vector_type(8))) float v8f;

#define LSTM_B    64
#define LSTM_T    1500
#define LSTM_H    128
#define LSTM_G    512            // 4*H
#define LSTM_M    (LSTM_B * LSTM_T)
#define GSTR      516            // padded LDS stride for gate buffer
#define HSTR      130            // padded LDS stride for h buffer
#define PRE_PF    12             // prefetch distance (steps) into L2

static __device__ __forceinline__ v8f wmma_f32(v2f a, v2f b, v8f c) {
  // D = A(16x4) * B(4x16) + C(16x16), all fp32
  return __builtin_amdgcn_wmma_f32_16x16x4_f32(
      /*neg_a=*/false, a, /*neg_b=*/false, b,
      /*c_mod=*/(short)0, c, /*reuse_a=*/false, /*reuse_b=*/false);
}

// CDNA5 hardware tanh (TRANS32). Inline-asm path adds a v_nop to satisfy the
// "1 independent op after TRANS before result use" rule (ISA 7.4) since the
// compiler does not model hazards through asm blocks.
static __device__ __forceinline__ float tanh_fast(float x) {
#if defined(__has_builtin) && __has_builtin(__builtin_amdgcn_tanhf)
  return __builtin_amdgcn_tanhf(x);
#else
  float y;
  asm volatile("v_tanh_f32 %0, %1\n\tv_nop" : "=v"(y) : "v"(x));
  return y;
#endif
}

static __device__ __forceinline__ float sigm(float x) {
  // sigmoid(x) = 0.5 * tanh(x/2) + 0.5  -> single TRANS op + 2 VALU
  return fmaf(0.5f, tanh_fast(0.5f * x), 0.5f);
}

// ---------------------------------------------------------------------------
// pre[d, m, 0:512] = sum_k in[m,k] * Wi[d, g, k]  + bi[d,g] + bh[d,g]
// grid = (M/16, 2 dirs), block = 256 (8 waves); wave w computes N-tiles
// {w, w+8, w+16, w+24} so the A fragment is reused 4x from registers.
// ---------------------------------------------------------------------------
__global__ __launch_bounds__(256) void lstm_pregemm(
    const float* __restrict__ in,   // [M, K]
    const float* __restrict__ Wi,   // [2, 512, K]
    const float* __restrict__ bi,   // [2, 512]
    const float* __restrict__ bh,   // [2, 512]
    float* __restrict__ pre,        // [2, M, 512]
    int K)
{
  const int m0   = blockIdx.x * 16;
  const int d    = blockIdx.y;
  const int lane = threadIdx.x & 31;
  const int wave = threadIdx.x >> 5;
  const int ln   = lane & 15;
  const int hi   = lane >> 4;      // 0: K pair {0,1}, 1: K pair {2,3}

  const float* W    = Wi  + (size_t)d * LSTM_G * K;
  float*       preD = pre + (size_t)d * LSTM_M * LSTM_G;

  const v8f vzero = {0.f, 0.f, 0.f, 0.f, 0.f, 0.f, 0.f, 0.f};
  v8f acc[4];
#pragma unroll
  for (int jj = 0; jj < 4; ++jj) acc[jj] = vzero;

  const float* arow = in + (size_t)(m0 + ln) * K + 2 * hi;
  const int ksteps = K >> 2;
  for (int s = 0; s < ksteps; ++s) {
    v2f a = *(const v2f*)(arow + 4 * s);
#pragma unroll
    for (int jj = 0; jj < 4; ++jj) {
      const int n0 = 16 * (wave + 8 * jj);
      v2f b = *(const v2f*)(W + (size_t)(n0 + ln) * K + 4 * s + 2 * hi);
      acc[jj] = wmma_f32(a, b, acc[jj]);
    }
  }

#pragma unroll
  for (int jj = 0; jj < 4; ++jj) {
    const int n0 = 16 * (wave + 8 * jj);
    const float bv = bi[d * LSTM_G + n0 + ln] + bh[d * LSTM_G + n0 + ln];
#pragma unroll
    for (int v = 0; v < 8; ++v) {
      const int row = v + (hi << 3);
      preD[(size_t)(m0 + row) * LSTM_G + n0 + ln] = acc[jj][v] + bv;
    }
  }
}

// ---------------------------------------------------------------------------
// Recurrent scan for one layer. grid = (4 batch tiles, 2 dirs),
// block = 1024 threads = 32 waves. Wave j owns gate columns [16j,16j+16).
// Whh^T fragments stay in registers across all T steps; the pre tile for
// step s+1 is loaded while step s computes (double buffer in registers).
// ---------------------------------------------------------------------------
__global__ __launch_bounds__(1024) void lstm_scan(
    const float* __restrict__ pre,   // [2, M, 512]
    const float* __restrict__ Whh,   // [2, 512, 128]
    float* __restrict__ out,         // [B, T, 256] (layer-0) or nullptr
    float* __restrict__ hfinal,      // [B, 256] (layer-1) or nullptr
    int T)
{
  __shared__ float gbuf[16 * GSTR];     // gate pre-activations, 16 x 512
  __shared__ float hbuf[16 * HSTR];     // h state, 16 x 128 (padded)
  __shared__ float cbuf[16 * 128];      // c state

  const int b0   = blockIdx.x * 16;
  const int d    = blockIdx.y;          // 0 = fwd, 1 = bwd
  const int tid  = threadIdx.x;
  const int lane = tid & 31;
  const int wave = tid >> 5;
  const int ln   = lane & 15;
  const int hi   = lane >> 4;
  const int n0   = wave * 16;           // this wave's gate-column base

  const float* preD = pre + (size_t)d * LSTM_M * LSTM_G;
  const float* Wd   = Whh + (size_t)d * LSTM_G * LSTM_H;

  // Load this wave's slice of Whh^T into registers: B[k][n] = Whh[n0+n][k].
  v2f bfrag[32];
#pragma unroll
  for (int s = 0; s < 32; ++s)
    bfrag[s] = *(const v2f*)(Wd + (size_t)(n0 + ln) * LSTM_H + 4 * s + 2 * hi);

  // h = c = 0
  for (int e = tid; e < 16 * HSTR; e += 1024) hbuf[e] = 0.0f;
  for (int e = tid; e < 16 * 128; e += 1024) cbuf[e] = 0.0f;
  __syncthreads();

  // Prime the pre-tile double buffer for the first step.
  float pnext[8];
  {
    const int t0 = d ? (T - 1) : 0;
#pragma unroll
    for (int v = 0; v < 8; ++v) {
      const int row = v + (hi << 3);
      pnext[v] = preD[((size_t)(b0 + row) * T + t0) * LSTM_G + n0 + ln];
    }
  }

  for (int s = 0; s < T; ++s) {
    const int t = d ? (T - 1 - s) : s;  // bwd direction walks time backwards

    // acc := this step's pre tile (loaded during the previous iteration)
    v8f acc;
#pragma unroll
    for (int v = 0; v < 8; ++v) acc[v] = pnext[v];

    // Issue next step's pre-tile loads now so they overlap the WMMA chain,
    // gate math and barriers of this step.
    if (s + 1 < T) {
      const int tn = d ? (T - 2 - s) : (s + 1);
#pragma unroll
      for (int v = 0; v < 8; ++v) {
        const int row = v + (hi << 3);
        pnext[v] = preD[((size_t)(b0 + row) * T + tn) * LSTM_G + n0 + ln];
      }
    }
    // Prefetch further ahead to stage HBM-resident pre tiles into L2.
    if (s + PRE_PF < T) {
      const int tp = d ? (T - 1 - (s + PRE_PF)) : (s + PRE_PF);
      __builtin_prefetch(&preD[((size_t)(b0 + ln) * T + tp) * LSTM_G + n0], 0, 0);
    }

    // acc += h @ Whh^T  (K = 128 -> 32 x wmma 16x16x4)
#pragma unroll
    for (int ss = 0; ss < 32; ++ss) {
      v2f a = *(const v2f*)(&hbuf[ln * HSTR + 4 * ss + 2 * hi]);
      acc = wmma_f32(a, bfrag[ss], acc);
    }

#pragma unroll
    for (int v = 0; v < 8; ++v) {
      const int row = v + (hi << 3);
      gbuf[row * GSTR + n0 + ln] = acc[v];
    }
    __syncthreads();

    // Gate math: 16 rows x 128 cols = 2048 elems, 2 per thread.
#pragma unroll
    for (int q = 0; q < 2; ++q) {
      const int e   = tid + q * 1024;
      const int row = e >> 7;
      const int col = e & 127;
      const float gi = gbuf[row * GSTR + col];
      const float gf = gbuf[row * GSTR + 128 + col];
      const float gc = gbuf[row * GSTR + 256 + col];
      const float go = gbuf[row * GSTR + 384 + col];
      const float c  = sigm(gf) * cbuf[row * 128 + col] +
                       sigm(gi) * tanh_fast(gc);
      const float h  = sigm(go) * tanh_fast(c);
      cbuf[row * 128 + col]  = c;
      hbuf[row * HSTR + col] = h;
      if (out)  // layer-0: both dirs write at true time index t (== reversal)
        out[((size_t)(b0 + row) * T + t) * 256 + d * 128 + col] = h;
    }
    __syncthreads();
  }

  if (hfinal) {
#pragma unroll
    for (int q = 0; q < 2; ++q) {
      const int e   = tid + q * 1024;
      const int row = e >> 7;
      const int col = e & 127;
      hfinal[(b0 + row) * 256 + d * 128 + col] = hbuf[row * HSTR + col];
    }
  }
}

// ---------------------------------------------------------------------------
extern "C" void kernel_launch(void* const* d_in, const int* in_sizes, int n_in,
                              void* d_out, int out_size, void* d_ws, size_t ws_size,
                              hipStream_t stream) {
  const float* x    = (const float*)d_in[0];  // [64,1500,80]
  const float* Wih0 = (const float*)d_in[1];  // [2,512,80]
  const float* Whh0 = (const float*)d_in[2];  // [2,512,128]
  const float* bih0 = (const float*)d_in[3];  // [2,512]
  const float* bhh0 = (const float*)d_in[4];
  const float* Wih1 = (const float*)d_in[5];  // [2,512,256]
  const float* Whh1 = (const float*)d_in[6];  // [2,512,128]
  const float* bih1 = (const float*)d_in[7];
  const float* bhh1 = (const float*)d_in[8];
  float* out = (float*)d_out;                 // [64, 256]

  float* pre  = (float*)d_ws;                           // 2*96000*512 f32
  float* out0 = pre + (size_t)2 * LSTM_M * LSTM_G;      // 96000*256   f32

  const dim3 gg(LSTM_M / 16, 2), gb(256);
  const dim3 sg(LSTM_B / 16, 2), sb(1024);

  // Layer 0
  lstm_pregemm<<<gg, gb, 0, stream>>>(x, Wih0, bih0, bhh0, pre, 80);
  lstm_scan<<<sg, sb, 0, stream>>>(pre, Whh0, out0, nullptr, LSTM_T);
  // Layer 1 (pre buffer reused)
  lstm_pregemm<<<gg, gb, 0, stream>>>(out0, Wih1, bih1, bhh1, pre, 256);
  lstm_scan<<<sg, sb, 0, stream>>>(pre, Whh1, nullptr, out, LSTM_T);
}